// RNNLayer_48026324304050
// MI455X (gfx1250) — compile-verified
//
#include <hip/hip_runtime.h>

// LSTM layer for MI455X (gfx1250): fp32 WMMA (V_WMMA_F32_16X16X4_F32), wave32.
// T=512, B=64, D=512, H=512.
//
// Phase 1: parallel x-projection GEMM, 2Mx2N register-blocked (1 load / WMMA).
// Phase 2: 512 sequential step kernels; each 16x16 gate-tile is split-K across
//          4 waves (chain of 128 WMMAs instead of 512), reduced via LDS in a
//          fixed order, then the LSTM cell update runs in registers.

typedef __attribute__((ext_vector_type(2))) float v2f;
typedef __attribute__((ext_vector_type(4))) float v4f;
typedef __attribute__((ext_vector_type(8))) float v8f;

#define T_LEN 512
#define B_SZ  64
#define D_SZ  512
#define H_SZ  512
#define G4H   (4 * H_SZ)

__device__ __forceinline__ float sigmoidf_fast(float x) {
  return 1.0f / (1.0f + __expf(-x));
}
__device__ __forceinline__ float tanhf_fast(float x) {
  float e = __expf(2.0f * x);
  return (e - 1.0f) / (e + 1.0f);
}

// ---------------------------------------------------------------------------
// Phase 1: xp[t*B+b, g] = sum_d x[t,b,d] * W_ih[g,d] + b_ih[g] + b_hh[g]
// M = T*B = 32768, N = 4H = 2048, K = D = 512.
// Each wave computes a 32x32 output block (2x2 tiles of 16x16):
// per K-chunk: 2 A loads + 2 B loads -> 4 WMMAs (1 load per WMMA).
// Super-tiles: 1024 (M) x 64 (N) = 65536 waves = 8192 blocks of 8 waves.
// ---------------------------------------------------------------------------
__global__ __launch_bounds__(256) void lstm_xproj_kernel(
    const float* __restrict__ x,    // [T*B, D]
    const float* __restrict__ Wih,  // [4H, D]
    const float* __restrict__ bih,  // [4H]
    const float* __restrict__ bhh,  // [4H]
    float* __restrict__ xp)         // [T*B, 4H]
{
  const int lane = threadIdx.x & 31;
  const int wave = blockIdx.x * 8 + (threadIdx.x >> 5);
  const int sn   = wave & 63;             // 64 N super-tiles (2048/32)
  const int sm   = wave >> 6;             // 1024 M super-tiles (32768/32)
  const int half = lane >> 4;             // 0 or 1
  const int l16  = lane & 15;
  const int koff = half * 2;              // K sub-offset per ISA A/B layout

  const int arow0 = sm * 32 + l16;        // global rows (t*B+b)
  const int arow1 = arow0 + 16;
  const int ncol0 = sn * 32 + l16;        // global gate columns
  const int ncol1 = ncol0 + 16;

  const float* ap0 = x   + (size_t)arow0 * D_SZ + koff;
  const float* ap1 = x   + (size_t)arow1 * D_SZ + koff;
  const float* wp0 = Wih + (size_t)ncol0 * D_SZ + koff;
  const float* wp1 = Wih + (size_t)ncol1 * D_SZ + koff;

  v8f acc00 = {}, acc01 = {}, acc10 = {}, acc11 = {};
#pragma unroll 4
  for (int k = 0; k < D_SZ; k += 4) {
    v2f a0 = *(const v2f*)(ap0 + k);
    v2f a1 = *(const v2f*)(ap1 + k);
    v2f b0 = *(const v2f*)(wp0 + k);
    v2f b1 = *(const v2f*)(wp1 + k);
    acc00 = __builtin_amdgcn_wmma_f32_16x16x4_f32(false, a0, false, b0, (short)0, acc00, false, false);
    acc01 = __builtin_amdgcn_wmma_f32_16x16x4_f32(false, a0, false, b1, (short)0, acc01, false, false);
    acc10 = __builtin_amdgcn_wmma_f32_16x16x4_f32(false, a1, false, b0, (short)0, acc10, false, false);
    acc11 = __builtin_amdgcn_wmma_f32_16x16x4_f32(false, a1, false, b1, (short)0, acc11, false, false);
  }

  // Fold both biases here so the recurrent kernel adds nothing.
  const float bb0 = bih[ncol0] + bhh[ncol0];
  const float bb1 = bih[ncol1] + bhh[ncol1];
  acc00 = acc00 + bb0;  acc10 = acc10 + bb0;
  acc01 = acc01 + bb1;  acc11 = acc11 + bb1;

  // D layout: VGPR r -> row M = r + half*8; lane's column = tile_base + l16.
  const int c0 = sn * 32 + l16;
  const int c1 = c0 + 16;
#pragma unroll
  for (int r = 0; r < 8; ++r) {
    const int m0 = sm * 32 + half * 8 + r;
    const int m1 = m0 + 16;
    xp[(size_t)m0 * G4H + c0] = acc00[r];
    xp[(size_t)m0 * G4H + c1] = acc01[r];
    xp[(size_t)m1 * G4H + c0] = acc10[r];
    xp[(size_t)m1 * G4H + c1] = acc11[r];
  }
}

// ---------------------------------------------------------------------------
// One recurrent step, split-K.
// Block = 256 threads = 8 waves = 2 gate-tiles x 4 K-slice waves.
// Grid = 64 blocks covers (B/16)*(H/16) = 128 tiles.
// Each K-slice wave runs a quarter of the K loop(s) (chain of 128 WMMAs),
// dumps 4 partial gate tiles to LDS, then the kw==0 wave reduces in fixed
// order, folds in xp (or biases), and applies the LSTM cell update.
// ---------------------------------------------------------------------------
__global__ __launch_bounds__(256) void lstm_step_kernel(
    const float* __restrict__ xt,    // [B, D]  (unused if use_xp)
    const float* __restrict__ xp_t,  // [B, 4H] (unused if !use_xp)
    const float* __restrict__ h_in,  // [B, H]
    const float* __restrict__ c_in,  // [B, H]
    const float* __restrict__ Wih,   // [4H, D]
    const float* __restrict__ Whh,   // [4H, H]
    const float* __restrict__ bih,   // [4H]
    const float* __restrict__ bhh,   // [4H]
    float* __restrict__ y_out,       // [B, H]
    float* __restrict__ c_out,       // [B, H]
    float* __restrict__ hT_out,      // [B, H] (written iff is_last)
    float* __restrict__ cT_out,      // [B, H] (written iff is_last)
    int use_xp, int is_last)
{
  // partials[tileInBlk][kw][gate][lane][r]  (r contiguous -> b128 LDS ops)
  __shared__ float lds[2 * 4 * 4 * 32 * 8];   // 32 KB

  const int lane      = threadIdx.x & 31;
  const int wave_blk  = threadIdx.x >> 5;     // 0..7
  const int tileInBlk = wave_blk >> 2;        // 0..1
  const int kw        = wave_blk & 3;         // K-slice id 0..3
  const int tile      = blockIdx.x * 2 + tileInBlk;   // 0..127
  const int tile_m    = tile & 3;             // 4 batch tiles
  const int tile_h    = tile >> 2;            // 32 hidden tiles
  const int half      = lane >> 4;
  const int l16       = lane & 15;
  const int koff      = half * 2;
  const int arow      = tile_m * 16 + l16;    // batch row for A operand
  const int gcol      = tile_h * 16 + l16;    // hidden unit column 0..511

  const int k_lo = kw * (D_SZ / 4);           // 128-wide K slice
  const int k_hi = k_lo + (D_SZ / 4);

  v8f acc_i = {}, acc_f = {}, acc_g = {}, acc_o = {};

  if (!use_xp) {
    // Fused input projection slice: gates += x_t @ W_ih^T  (K in [k_lo,k_hi))
    const float* ap = xt  + (size_t)arow * D_SZ + koff;
    const float* wi = Wih + (size_t)(0 * H_SZ + gcol) * D_SZ + koff;
    const float* wf = Wih + (size_t)(1 * H_SZ + gcol) * D_SZ + koff;
    const float* wg = Wih + (size_t)(2 * H_SZ + gcol) * D_SZ + koff;
    const float* wo = Wih + (size_t)(3 * H_SZ + gcol) * D_SZ + koff;
#pragma unroll 2
    for (int k = k_lo; k < k_hi; k += 4) {
      v2f a  = *(const v2f*)(ap + k);
      v2f bi = *(const v2f*)(wi + k);
      v2f bf = *(const v2f*)(wf + k);
      v2f bg = *(const v2f*)(wg + k);
      v2f bo = *(const v2f*)(wo + k);
      acc_i = __builtin_amdgcn_wmma_f32_16x16x4_f32(false, a, false, bi, (short)0, acc_i, false, false);
      acc_f = __builtin_amdgcn_wmma_f32_16x16x4_f32(false, a, false, bf, (short)0, acc_f, false, false);
      acc_g = __builtin_amdgcn_wmma_f32_16x16x4_f32(false, a, false, bg, (short)0, acc_g, false, false);
      acc_o = __builtin_amdgcn_wmma_f32_16x16x4_f32(false, a, false, bo, (short)0, acc_o, false, false);
    }
  }

  {
    // Recurrent slice: gates += h_{t-1} @ W_hh^T  (K in [k_lo,k_hi))
    const float* ap = h_in + (size_t)arow * H_SZ + koff;
    const float* wi = Whh + (size_t)(0 * H_SZ + gcol) * H_SZ + koff;
    const float* wf = Whh + (size_t)(1 * H_SZ + gcol) * H_SZ + koff;
    const float* wg = Whh + (size_t)(2 * H_SZ + gcol) * H_SZ + koff;
    const float* wo = Whh + (size_t)(3 * H_SZ + gcol) * H_SZ + koff;
#pragma unroll 2
    for (int k = k_lo; k < k_hi; k += 4) {
      v2f a  = *(const v2f*)(ap + k);
      v2f bi = *(const v2f*)(wi + k);
      v2f bf = *(const v2f*)(wf + k);
      v2f bg = *(const v2f*)(wg + k);
      v2f bo = *(const v2f*)(wo + k);
      acc_i = __builtin_amdgcn_wmma_f32_16x16x4_f32(false, a, false, bi, (short)0, acc_i, false, false);
      acc_f = __builtin_amdgcn_wmma_f32_16x16x4_f32(false, a, false, bf, (short)0, acc_f, false, false);
      acc_g = __builtin_amdgcn_wmma_f32_16x16x4_f32(false, a, false, bg, (short)0, acc_g, false, false);
      acc_o = __builtin_amdgcn_wmma_f32_16x16x4_f32(false, a, false, bo, (short)0, acc_o, false, false);
    }
  }

  // Dump partial gate tiles to LDS: 8 contiguous floats per (gate,lane).
  {
    float* base = &lds[(size_t)((tileInBlk * 4 + kw) * 4) * 32 * 8 + lane * 8];
    v8f* accs[4] = {&acc_i, &acc_f, &acc_g, &acc_o};
#pragma unroll
    for (int g = 0; g < 4; ++g) {
      float* p = base + g * 32 * 8;
      v8f a = *accs[g];
      *(v4f*)(p)     = __builtin_shufflevector(a, a, 0, 1, 2, 3);
      *(v4f*)(p + 4) = __builtin_shufflevector(a, a, 4, 5, 6, 7);
    }
  }

  __syncthreads();

  if (kw == 0) {
    // Fixed-order split-K reduction (deterministic).
    float gs[4][8];
#pragma unroll
    for (int g = 0; g < 4; ++g) {
      v4f lo = {}, hi = {};
#pragma unroll
      for (int w = 0; w < 4; ++w) {
        const float* p =
            &lds[(size_t)((tileInBlk * 4 + w) * 4 + g) * 32 * 8 + lane * 8];
        lo += *(const v4f*)(p);
        hi += *(const v4f*)(p + 4);
      }
#pragma unroll
      for (int r = 0; r < 4; ++r) { gs[g][r] = lo[r]; gs[g][4 + r] = hi[r]; }
    }

    if (use_xp) {
      // x-projection (biases already folded in) from the phase-1 buffer.
#pragma unroll
      for (int r = 0; r < 8; ++r) {
        const int m = tile_m * 16 + half * 8 + r;
        const float* row = xp_t + (size_t)m * G4H;
        gs[0][r] += row[0 * H_SZ + gcol];
        gs[1][r] += row[1 * H_SZ + gcol];
        gs[2][r] += row[2 * H_SZ + gcol];
        gs[3][r] += row[3 * H_SZ + gcol];
      }
    } else {
#pragma unroll
      for (int g = 0; g < 4; ++g) {
        const float bb = bih[g * H_SZ + gcol] + bhh[g * H_SZ + gcol];
#pragma unroll
        for (int r = 0; r < 8; ++r) gs[g][r] += bb;
      }
    }

    // LSTM cell update in registers.
#pragma unroll
    for (int r = 0; r < 8; ++r) {
      const int m  = tile_m * 16 + half * 8 + r;   // batch row
      const int hh = tile_h * 16 + l16;            // hidden column
      const float ig = sigmoidf_fast(gs[0][r]);
      const float fg = sigmoidf_fast(gs[1][r]);
      const float gg = tanhf_fast(gs[2][r]);
      const float og = sigmoidf_fast(gs[3][r]);
      const float cp = c_in[(size_t)m * H_SZ + hh];
      const float cn = fg * cp + ig * gg;
      const float hn = og * tanhf_fast(cn);
      y_out[(size_t)m * H_SZ + hh] = hn;
      c_out[(size_t)m * H_SZ + hh] = cn;
      if (is_last) {
        hT_out[(size_t)m * H_SZ + hh] = hn;
        cT_out[(size_t)m * H_SZ + hh] = cn;
      }
    }
  }
}

// ---------------------------------------------------------------------------
extern "C" void kernel_launch(void* const* d_in, const int* in_sizes, int n_in,
                              void* d_out, int out_size, void* d_ws, size_t ws_size,
                              hipStream_t stream) {
  (void)in_sizes; (void)n_in; (void)out_size;

  const float* x   = (const float*)d_in[0];  // [T, B, D]
  const float* h0  = (const float*)d_in[1];  // [B, H]
  const float* c0  = (const float*)d_in[2];  // [B, H]
  const float* Wih = (const float*)d_in[3];  // [4H, D]
  const float* Whh = (const float*)d_in[4];  // [4H, H]
  const float* bih = (const float*)d_in[5];  // [4H]
  const float* bhh = (const float*)d_in[6];  // [4H]

  float* y  = (float*)d_out;                          // [T, B, H]
  float* hT = y + (size_t)T_LEN * B_SZ * H_SZ;        // [B, H]
  float* cT = hT + (size_t)B_SZ * H_SZ;               // [B, H]

  // Workspace: c ping buffer (128 KB) + optional x_proj (256 MB).
  float* cbuf = (float*)d_ws;
  const size_t cbuf_bytes = (size_t)B_SZ * H_SZ * sizeof(float);
  const size_t xp_bytes   = (size_t)T_LEN * B_SZ * G4H * sizeof(float);
  float* xp = (float*)((char*)d_ws + cbuf_bytes);
  const int use_xp = (ws_size >= cbuf_bytes + xp_bytes) ? 1 : 0;

  if (use_xp) {
    // Parallel input-projection GEMM: 65536 waves (32x32 blocks), 8/block.
    lstm_xproj_kernel<<<8192, 256, 0, stream>>>(x, Wih, bih, bhh, xp);
  }

  // Sequential recurrence: y[t] doubles as the h-state history.
  for (int t = 0; t < T_LEN; ++t) {
    const float* xt   = x + (size_t)t * B_SZ * D_SZ;
    const float* xp_t = xp + (size_t)t * B_SZ * G4H;
    const float* h_in = (t == 0) ? h0 : (y + (size_t)(t - 1) * B_SZ * H_SZ);
    const float* c_in = (t == 0) ? c0 : cbuf;
    float* y_out = y + (size_t)t * B_SZ * H_SZ;
    const int is_last = (t == T_LEN - 1) ? 1 : 0;
    lstm_step_kernel<<<64, 256, 0, stream>>>(xt, xp_t, h_in, c_in,
                                             Wih, Whh, bih, bhh,
                                             y_out, cbuf, hT, cT,
                                             use_xp, is_last);
  }
}